// MyModel_61933428409043
// MI455X (gfx1250) — compile-verified
//
#include <hip/hip_runtime.h>

typedef __attribute__((ext_vector_type(2))) float v2f;
typedef __attribute__((ext_vector_type(8))) float v8f;

#define HID   51
#define TIN   512
#define BATCH 2048
#define NP    256   // padded gate columns (4 gates x 64)
#define KP    64    // padded hidden width (LDS row stride)
#define ROWS  16    // batch rows per workgroup (WMMA M)
#define NKK   13    // K-steps of 4 -> covers K=52 >= 51

__device__ __forceinline__ float sigf(float x) {
    return 1.0f / (1.0f + __expf(-x));
}
__device__ __forceinline__ float tanhf_(float x) {
    // tanh(x) = 1 - 2/(e^{2x}+1)
    return 1.0f - 2.0f / (__expf(2.0f * x) + 1.0f);
}

// Load one B-operand pair for V_WMMA_F32_16X16X4_F32:
// lane (l<16): rows K=k0,k0+1 ; lane (l>=16): rows K=k0+2,k0+3 (k0 passed in
// already includes the +2*hi). Weights are [204][51] row-major (W[n][k]);
// B[k][n] = W[n][k]. Out-of-range (padded) entries are zero.
__device__ __forceinline__ v2f ldw(const float* __restrict__ W, int norig,
                                   int k0, bool vn) {
    int i0 = (vn && k0 < HID) ? (norig * HID + k0) : 0;
    int i1 = (vn && (k0 + 1) < HID) ? (norig * HID + k0 + 1) : 0;
    float a = W[i0];
    float b = W[i1];
    v2f r;
    r.x = (vn && k0 < HID) ? a : 0.0f;
    r.y = (vn && (k0 + 1) < HID) ? b : 0.0f;
    return r;
}

__global__ __launch_bounds__(512) void lstm2_wmma_kernel(
    const float* __restrict__ input,   // [B, TIN]
    const float* __restrict__ Wih1,    // [204, 1]
    const float* __restrict__ Whh1,    // [204, 51]
    const float* __restrict__ bih1, const float* __restrict__ bhh1,
    const float* __restrict__ Wih2,    // [204, 51]
    const float* __restrict__ Whh2,    // [204, 51]
    const float* __restrict__ bih2, const float* __restrict__ bhh2,
    const float* __restrict__ Wlin,    // [1, 51]
    const float* __restrict__ blin,    // [1]
    float* __restrict__ out,           // [B, Tout]
    int Tout)
{
    __shared__ float sh1[ROWS * KP];     // layer-1 hidden state (zero padded)
    __shared__ float sh2[ROWS * KP];     // layer-2 hidden state (zero padded)
    __shared__ float sg[NP * ROWS];      // gate scratch, column-major [n][row]
    __shared__ float sb1[NP], sb2[NP], swi1[NP];
    __shared__ float swl[KP];
    __shared__ float sx[ROWS];

    const int tid  = threadIdx.x;
    const int lane = tid & 31;
    const int wv   = tid >> 5;           // wave id 0..15 -> N-tile
    const int hi   = lane >> 4;          // half-wave selector
    const int lrow = lane & 15;
    const int n    = (wv << 4) + lrow;   // gate column owned by this lane
    const int b0   = blockIdx.x * ROWS;

    // ---------------- init ----------------
    for (int i = tid; i < ROWS * KP; i += 512) { sh1[i] = 0.0f; sh2[i] = 0.0f; }
    if (tid < NP) {
        int g = tid >> 6, j = tid & 63;
        bool v = j < HID;
        int o = g * HID + j;
        sb1[tid]  = v ? (bih1[o] + bhh1[o]) : 0.0f;
        sb2[tid]  = v ? (bih2[o] + bhh2[o]) : 0.0f;
        swi1[tid] = v ? Wih1[o] : 0.0f;
    }
    if (tid < KP)   swl[tid] = (tid < HID) ? Wlin[tid] : 0.0f;
    if (tid < ROWS) sx[tid]  = 0.0f;

    // ---- pin all weight B-tiles in VGPRs (78 VGPRs / wave) ----
    const int  gN    = n >> 6;
    const int  jN    = n & 63;
    const bool vn    = jN < HID;
    const int  norig = gN * HID + jN;
    v2f wB1[NKK], wBi2[NKK], wBh2[NKK];
#pragma unroll
    for (int kk = 0; kk < NKK; ++kk) {
        int k0 = 4 * kk + 2 * hi;
        wB1[kk]  = ldw(Whh1, norig, k0, vn);
        wBi2[kk] = ldw(Wih2, norig, k0, vn);
        wBh2[kk] = ldw(Whh2, norig, k0, vn);
    }

    float c1v[2] = {0.0f, 0.0f};
    float c2v[2] = {0.0f, 0.0f};
    const float bl = blin[0];

    const float* ha1 = sh1 + (lrow << 6) + (hi << 1);  // A-operand base, layer-1 h
    const float* ha2 = sh2 + (lrow << 6) + (hi << 1);  // A-operand base, layer-2 h
    float* gp = sg + n * ROWS + (hi << 3);             // C-tile dump base

    __syncthreads();

    for (int t = 0; t < Tout; ++t) {
        // ---------- G1: gates1 = h1 @ W_hh1^T (two interleaved acc chains) ----
        if (tid < ROWS && t < TIN)
            sx[tid] = input[(b0 + tid) * TIN + t];

        v8f accA = {};
        v8f accB = {};
#pragma unroll
        for (int kk = 0; kk < NKK; kk += 2) {
            v2f a = *(const v2f*)(ha1 + 4 * kk);
            accA = __builtin_amdgcn_wmma_f32_16x16x4_f32(
                false, a, false, wB1[kk], (short)0, accA, false, false);
            if (kk + 1 < NKK) {
                v2f b = *(const v2f*)(ha1 + 4 * (kk + 1));
                accB = __builtin_amdgcn_wmma_f32_16x16x4_f32(
                    false, b, false, wB1[kk + 1], (short)0, accB, false, false);
            }
        }
        {
            v8f acc = accA + accB;
            *(float4*)(gp)     = make_float4(acc[0], acc[1], acc[2], acc[3]);
            *(float4*)(gp + 4) = make_float4(acc[4], acc[5], acc[6], acc[7]);
        }
        __syncthreads();

        // ---------- U1: LSTM cell 1 (c1 in registers, h1 -> LDS) ----------
#pragma unroll
        for (int e = 0; e < 2; ++e) {
            int idx = tid + (e << 9);
            int row = idx >> 6;
            int jj  = idx & 63;
            if (jj < HID) {
                float x  = sx[row];
                float gi = sg[jj * ROWS + row]          + sb1[jj]        + x * swi1[jj];
                float gf = sg[(64 + jj) * ROWS + row]   + sb1[64 + jj]   + x * swi1[64 + jj];
                float gg = sg[(128 + jj) * ROWS + row]  + sb1[128 + jj]  + x * swi1[128 + jj];
                float go = sg[(192 + jj) * ROWS + row]  + sb1[192 + jj]  + x * swi1[192 + jj];
                float c  = sigf(gf) * c1v[e] + sigf(gi) * tanhf_(gg);
                c1v[e]   = c;
                sh1[(row << 6) + jj] = sigf(go) * tanhf_(c);
            }
        }
        __syncthreads();

        // ---------- G2: gates2 = h1 @ W_ih2^T + h2 @ W_hh2^T ----------
        // Two naturally independent matmuls -> two interleaved acc chains.
        accA = (v8f){};
        accB = (v8f){};
#pragma unroll
        for (int kk = 0; kk < NKK; ++kk) {
            v2f a = *(const v2f*)(ha1 + 4 * kk);
            accA = __builtin_amdgcn_wmma_f32_16x16x4_f32(
                false, a, false, wBi2[kk], (short)0, accA, false, false);
            v2f b = *(const v2f*)(ha2 + 4 * kk);
            accB = __builtin_amdgcn_wmma_f32_16x16x4_f32(
                false, b, false, wBh2[kk], (short)0, accB, false, false);
        }
        {
            v8f acc = accA + accB;
            *(float4*)(gp)     = make_float4(acc[0], acc[1], acc[2], acc[3]);
            *(float4*)(gp + 4) = make_float4(acc[4], acc[5], acc[6], acc[7]);
        }
        __syncthreads();

        // ---------- U2: LSTM cell 2 (c2 in registers, h2 -> LDS) ----------
#pragma unroll
        for (int e = 0; e < 2; ++e) {
            int idx = tid + (e << 9);
            int row = idx >> 6;
            int jj  = idx & 63;
            if (jj < HID) {
                float gi = sg[jj * ROWS + row]          + sb2[jj];
                float gf = sg[(64 + jj) * ROWS + row]   + sb2[64 + jj];
                float gg = sg[(128 + jj) * ROWS + row]  + sb2[128 + jj];
                float go = sg[(192 + jj) * ROWS + row]  + sb2[192 + jj];
                float c  = sigf(gf) * c2v[e] + sigf(gi) * tanhf_(gg);
                c2v[e]   = c;
                sh2[(row << 6) + jj] = sigf(go) * tanhf_(c);
            }
        }
        __syncthreads();

        // ---------- OUT: y = h2 @ W_lin^T + b (16-lane shuffle reduce) ----------
        if (tid < 256) {
            int row = tid >> 4, s = tid & 15;
            const float* h = sh2 + (row << 6);
            float p = h[s]      * swl[s]
                    + h[s + 16] * swl[s + 16]
                    + h[s + 32] * swl[s + 32]
                    + h[s + 48] * swl[s + 48];   // padded entries are zero
            p += __shfl_xor(p, 8, 32);
            p += __shfl_xor(p, 4, 32);
            p += __shfl_xor(p, 2, 32);
            p += __shfl_xor(p, 1, 32);
            if (s == 0) {
                float o = p + bl;
                out[(b0 + row) * Tout + t] = o;
                if (t >= TIN - 1) sx[row] = o;  // autoregressive feed for t>=TIN
            }
        }
        // no barrier needed: next G1 reads sh1 (sealed since U1 barrier) and
        // writes sg/sx disjointly from OUT's reads/writes.
    }
}

extern "C" void kernel_launch(void* const* d_in, const int* in_sizes, int n_in,
                              void* d_out, int out_size, void* d_ws, size_t ws_size,
                              hipStream_t stream) {
    const float* input = (const float*)d_in[0];
    const float* Wih1  = (const float*)d_in[1];
    const float* Whh1  = (const float*)d_in[2];
    const float* bih1  = (const float*)d_in[3];
    const float* bhh1  = (const float*)d_in[4];
    const float* Wih2  = (const float*)d_in[5];
    const float* Whh2  = (const float*)d_in[6];
    const float* bih2  = (const float*)d_in[7];
    const float* bhh2  = (const float*)d_in[8];
    const float* Wlin  = (const float*)d_in[9];
    const float* blin  = (const float*)d_in[10];
    // d_in[11] is `future`; Tout is recovered from out_size (B * (TIN+future)).
    int Tout = out_size / BATCH;

    dim3 grid(BATCH / ROWS);   // 128 workgroups
    dim3 block(512);           // 16 wave32s: one N-tile of gate columns each
    lstm2_wmma_kernel<<<grid, block, 0, stream>>>(
        input, Wih1, Whh1, bih1, bhh1, Wih2, Whh2, bih2, bhh2,
        Wlin, blin, (float*)d_out, Tout);
}